// GNN_89910845374716
// MI455X (gfx1250) — compile-verified
//
#include <hip/hip_runtime.h>

typedef __attribute__((ext_vector_type(2))) float v2f;
typedef __attribute__((ext_vector_type(8))) float v8f;

#define N_NODES  100000
#define N_EDGES  3200000
#define IN_DIM   128
#define HID1     32
#define HID2     16
#define N_GRAPHS 512
#define M_TILES  (N_NODES / 16)   // 6250 exactly
#define W1_STRIDE 130             // padded transposed row stride (avoids LDS bank conflicts)
#define W2_STRIDE 34

// ---------------------------------------------------------------- init
__global__ void k_init(float* __restrict__ deg, float* __restrict__ pooled,
                       float* __restrict__ counts) {
    int i = blockIdx.x * 256 + threadIdx.x;
    if (i < N_NODES) deg[i] = 1.0f;               // self-loop contributes 1
    if (i < N_GRAPHS * HID2) pooled[i] = 0.0f;
    if (i < N_GRAPHS) counts[i] = 0.0f;
}

// ---------------------------------------------------------------- degree
__global__ void k_degree(const int* __restrict__ dst, float* __restrict__ deg) {
    int e = blockIdx.x * 256 + threadIdx.x;
    if (e < N_EDGES) atomicAdd(&deg[dst[e]], 1.0f);
}

__global__ void k_rsqrt(float* __restrict__ deg) {
    int i = blockIdx.x * 256 + threadIdx.x;
    if (i < N_NODES) deg[i] = rsqrtf(deg[i]);     // deg >= 1 always
}

// ---------------------------------------------------------------- GEMM1: h1 = x @ W1  [N,128]x[128,32]
__global__ void __launch_bounds__(256)
k_gemm1(const float* __restrict__ x, const float* __restrict__ W,
        float* __restrict__ h1) {
    // W staged TRANSPOSED with padded stride: Wt[n][k], so a B fragment
    // (W[kb][n], W[kb+1][n]) is one contiguous 8B LDS read (ds_load_b64).
    __shared__ float Wt[HID1 * W1_STRIDE];        // 16.6 KB
    int t = threadIdx.x;
    for (int i = t; i < IN_DIM * HID1; i += 256) {
        int k = i >> 5, n = i & 31;               // W is [128,32] row-major
        Wt[n * W1_STRIDE + k] = W[i];
    }
    __syncthreads();

    int wave = t >> 5, lane = t & 31;
    int tile = blockIdx.x * 8 + wave;
    if (tile >= M_TILES) return;                  // wave-uniform exit

    int m0    = tile * 16;
    int mrow  = m0 + (lane & 15);
    int khalf = (lane >> 4) << 1;                 // 0 for lanes 0-15, 2 for 16-31
    int ncol  = lane & 15;
    const float* xrow = x + (size_t)mrow * IN_DIM + khalf;
    const float* brow0 = Wt + ncol * W1_STRIDE + khalf;
    const float* brow1 = Wt + (ncol + 16) * W1_STRIDE + khalf;

    v8f c0 = {}; v8f c1 = {};
    #pragma unroll
    for (int k = 0; k < IN_DIM; k += 4) {
        v2f a  = *(const v2f*)(xrow + k);         // A[m][kb..kb+1]
        v2f b0 = *(const v2f*)(brow0 + k);        // B[kb..kb+1][n]      (n = 0..15)
        v2f b1 = *(const v2f*)(brow1 + k);        // B[kb..kb+1][n+16]
        c0 = __builtin_amdgcn_wmma_f32_16x16x4_f32(false, a, false, b0,
                                                   (short)0, c0, false, false);
        c1 = __builtin_amdgcn_wmma_f32_16x16x4_f32(false, a, false, b1,
                                                   (short)0, c1, false, false);
    }
    int rbase = m0 + ((lane >> 4) << 3);          // M = r (+8 for upper half-wave)
    #pragma unroll
    for (int r = 0; r < 8; ++r) {
        h1[(size_t)(rbase + r) * HID1 + ncol]      = c0[r];
        h1[(size_t)(rbase + r) * HID1 + 16 + ncol] = c1[r];
    }
}

// ---------------------------------------------------------------- GEMM2: h2 = a1 @ W2  [N,32]x[32,16]
__global__ void __launch_bounds__(256)
k_gemm2(const float* __restrict__ a1, const float* __restrict__ W,
        float* __restrict__ h2) {
    __shared__ float Wt[HID2 * W2_STRIDE];        // 2.2 KB, transposed+padded
    int t = threadIdx.x;
    for (int i = t; i < HID1 * HID2; i += 256) {
        int k = i >> 4, n = i & 15;               // W is [32,16] row-major
        Wt[n * W2_STRIDE + k] = W[i];
    }
    __syncthreads();

    int wave = t >> 5, lane = t & 31;
    int tile = blockIdx.x * 8 + wave;
    if (tile >= M_TILES) return;

    int m0    = tile * 16;
    int mrow  = m0 + (lane & 15);
    int khalf = (lane >> 4) << 1;
    int ncol  = lane & 15;
    const float* arow = a1 + (size_t)mrow * HID1 + khalf;
    const float* brow = Wt + ncol * W2_STRIDE + khalf;

    v8f c = {};
    #pragma unroll
    for (int k = 0; k < HID1; k += 4) {
        v2f a = *(const v2f*)(arow + k);
        v2f b = *(const v2f*)(brow + k);
        c = __builtin_amdgcn_wmma_f32_16x16x4_f32(false, a, false, b,
                                                  (short)0, c, false, false);
    }
    int rbase = m0 + ((lane >> 4) << 3);
    #pragma unroll
    for (int r = 0; r < 8; ++r)
        h2[(size_t)(rbase + r) * HID2 + ncol] = c[r];
}

// ---------------------------------------------------------------- self-loop: agg = h * dinv^2 (also zero-init)
__global__ void k_selfloop32(const float* __restrict__ h, const float* __restrict__ dinv,
                             float* __restrict__ agg) {
    int idx = blockIdx.x * 256 + threadIdx.x;     // N*8 float4 groups
    if (idx >= N_NODES * 8) return;
    int node = idx >> 3, g = idx & 7;
    float w = dinv[node]; w *= w;
    float4 v = ((const float4*)(h + (size_t)node * HID1))[g];
    v.x *= w; v.y *= w; v.z *= w; v.w *= w;
    ((float4*)(agg + (size_t)node * HID1))[g] = v;
}

__global__ void k_selfloop16(const float* __restrict__ h, const float* __restrict__ dinv,
                             float* __restrict__ agg) {
    int idx = blockIdx.x * 256 + threadIdx.x;     // N*4 float4 groups
    if (idx >= N_NODES * 4) return;
    int node = idx >> 2, g = idx & 3;
    float w = dinv[node]; w *= w;
    float4 v = ((const float4*)(h + (size_t)node * HID2))[g];
    v.x *= w; v.y *= w; v.z *= w; v.w *= w;
    ((float4*)(agg + (size_t)node * HID2))[g] = v;
}

// ---------------------------------------------------------------- edge scatter (layer 1, 32 ch, 8 thr/edge)
__global__ void k_edge32(const int* __restrict__ src, const int* __restrict__ dst,
                         const float* __restrict__ dinv, const float* __restrict__ h1,
                         float* __restrict__ agg) {
    int tid = blockIdx.x * 256 + threadIdx.x;     // N_EDGES*8 = 25.6M
    if (tid >= N_EDGES * 8) return;
    int e = tid >> 3, g = tid & 7;
    int s = src[e], d = dst[e];
    float nrm = dinv[s] * dinv[d];
    float4 v = ((const float4*)(h1 + (size_t)s * HID1))[g];
    float* p = agg + (size_t)d * HID1 + g * 4;
    atomicAdd(p + 0, v.x * nrm); atomicAdd(p + 1, v.y * nrm);
    atomicAdd(p + 2, v.z * nrm); atomicAdd(p + 3, v.w * nrm);
}

// ---------------------------------------------------------------- edge scatter (layer 2, 16 ch, 4 thr/edge)
__global__ void k_edge16(const int* __restrict__ src, const int* __restrict__ dst,
                         const float* __restrict__ dinv, const float* __restrict__ h2,
                         float* __restrict__ agg) {
    int tid = blockIdx.x * 256 + threadIdx.x;     // N_EDGES*4 = 12.8M
    if (tid >= N_EDGES * 4) return;
    int e = tid >> 2, g = tid & 3;
    int s = src[e], d = dst[e];
    float nrm = dinv[s] * dinv[d];
    float4 v = ((const float4*)(h2 + (size_t)s * HID2))[g];
    float* p = agg + (size_t)d * HID2 + g * 4;
    atomicAdd(p + 0, v.x * nrm); atomicAdd(p + 1, v.y * nrm);
    atomicAdd(p + 2, v.z * nrm); atomicAdd(p + 3, v.w * nrm);
}

// ---------------------------------------------------------------- bias + relu (layer 1, in place)
__global__ void k_relubias32(float* __restrict__ agg, const float* __restrict__ b) {
    int idx = blockIdx.x * 256 + threadIdx.x;
    if (idx >= N_NODES * HID1) return;
    float v = agg[idx] + b[idx & (HID1 - 1)];
    agg[idx] = v > 0.0f ? v : 0.0f;
}

// ---------------------------------------------------------------- bias + relu + mean-pool accumulate
__global__ void k_pool(const float* __restrict__ agg2, const float* __restrict__ b2,
                       const int* __restrict__ batch, float* __restrict__ pooled,
                       float* __restrict__ counts) {
    int tid = blockIdx.x * 256 + threadIdx.x;     // N*4 float4 groups
    if (tid >= N_NODES * 4) return;
    int node = tid >> 2, g = tid & 3;
    int gr = batch[node];
    float4 v  = ((const float4*)(agg2 + (size_t)node * HID2))[g];
    float4 bb = ((const float4*)b2)[g];
    float r0 = fmaxf(v.x + bb.x, 0.0f), r1 = fmaxf(v.y + bb.y, 0.0f);
    float r2 = fmaxf(v.z + bb.z, 0.0f), r3 = fmaxf(v.w + bb.w, 0.0f);
    float* p = pooled + (size_t)gr * HID2 + g * 4;
    atomicAdd(p + 0, r0); atomicAdd(p + 1, r1);
    atomicAdd(p + 2, r2); atomicAdd(p + 3, r3);
    if (g == 0) atomicAdd(&counts[gr], 1.0f);
}

// ---------------------------------------------------------------- final: out = (pooled/counts) @ W3 + b3
__global__ void k_final(const float* __restrict__ pooled, const float* __restrict__ counts,
                        const float* __restrict__ W3, const float* __restrict__ b3,
                        float* __restrict__ out) {
    int g = blockIdx.x * 256 + threadIdx.x;
    if (g >= N_GRAPHS) return;
    float c = fmaxf(counts[g], 1.0f);
    float s = 0.0f;
    #pragma unroll
    for (int i = 0; i < HID2; ++i) s += pooled[g * HID2 + i] * W3[i];
    out[g] = s / c + b3[0];
}

// ---------------------------------------------------------------- host
extern "C" void kernel_launch(void* const* d_in, const int* in_sizes, int n_in,
                              void* d_out, int out_size, void* d_ws, size_t ws_size,
                              hipStream_t stream) {
    const float* x     = (const float*)d_in[0];
    const int*   ei    = (const int*)d_in[1];      // [2, E] flat: row0=src, row1=dst
    const int*   batch = (const int*)d_in[2];
    const float* W1    = (const float*)d_in[3];
    const float* b1    = (const float*)d_in[4];
    const float* W2    = (const float*)d_in[5];
    const float* b2    = (const float*)d_in[6];
    const float* W3    = (const float*)d_in[7];
    const float* b3    = (const float*)d_in[8];
    float*       out   = (float*)d_out;
    const int* src = ei;
    const int* dst = ei + N_EDGES;

    // workspace layout (floats, 256B-aligned segments)
    float* ws     = (float*)d_ws;
    float* dinv   = ws;                         // N (deg, then rsqrt in place)
    float* h1     = dinv + 100352;              // N*32
    float* agg1   = h1   + N_NODES * HID1;      // N*32 (becomes a1 after relu)
    float* h2     = agg1 + N_NODES * HID1;      // N*16
    float* agg2   = h2   + N_NODES * HID2;      // N*16
    float* pooled = agg2 + N_NODES * HID2;      // 512*16
    float* counts = pooled + N_GRAPHS * HID2;   // 512
    (void)in_sizes; (void)n_in; (void)out_size; (void)ws_size;

    k_init      <<<(N_NODES + 255) / 256, 256, 0, stream>>>(dinv, pooled, counts);
    k_degree    <<<(N_EDGES + 255) / 256, 256, 0, stream>>>(dst, dinv);
    k_rsqrt     <<<(N_NODES + 255) / 256, 256, 0, stream>>>(dinv);

    k_gemm1     <<<(M_TILES + 7) / 8, 256, 0, stream>>>(x, W1, h1);
    k_selfloop32<<<(N_NODES * 8 + 255) / 256, 256, 0, stream>>>(h1, dinv, agg1);
    k_edge32    <<<(N_EDGES * 8 + 255) / 256, 256, 0, stream>>>(src, dst, dinv, h1, agg1);
    k_relubias32<<<(N_NODES * HID1 + 255) / 256, 256, 0, stream>>>(agg1, b1);

    k_gemm2     <<<(M_TILES + 7) / 8, 256, 0, stream>>>(agg1, W2, h2);
    k_selfloop16<<<(N_NODES * 4 + 255) / 256, 256, 0, stream>>>(h2, dinv, agg2);
    k_edge16    <<<(N_EDGES * 4 + 255) / 256, 256, 0, stream>>>(src, dst, dinv, h2, agg2);

    k_pool      <<<(N_NODES * 4 + 255) / 256, 256, 0, stream>>>(agg2, b2, batch, pooled, counts);
    k_final     <<<(N_GRAPHS + 255) / 256, 256, 0, stream>>>(pooled, counts, W3, b3, out);
}